// Patches_2705829396791
// MI455X (gfx1250) — compile-verified
//
#include <hip/hip_runtime.h>
#include <hip/hip_bf16.h>
#include <stdint.h>

// ---------------- problem constants ----------------
#define Pz      7
#define NPq     10
#define WIN     20
#define STRIDE  4
#define Lc      14          // WIN - P + 1
#define HALF    7           // WIN/2 - P/2
#define Hc      160
#define Wc      160
#define PLANE   (Hc*Wc)     // 25600
#define BCc     32          // B*C
#define NX      36          // positions per axis
#define NPOS    (NX*NX)     // 1296
#define NPATCH  (NPOS*BCc*NPq)   // 414720 patch vectors
#define PATCHV  49
#define TILES   (NPATCH/16)      // 25920
#define NSTEP   (NPOS*NPq)       // 12960 sequential steps per plane

typedef __attribute__((ext_vector_type(2))) float v2f;
typedef __attribute__((ext_vector_type(8))) float v8f;
typedef __attribute__((ext_vector_type(4))) int v4i;
typedef __attribute__((ext_vector_type(4))) unsigned int u32x4;
typedef __attribute__((ext_vector_type(8))) int i32x8;
typedef __attribute__((ext_vector_type(4))) int i32x4;

// ---- gfx1250 feature guards (host pass sees the fallbacks) ----
#if defined(__AMDGCN__) && __has_builtin(__builtin_amdgcn_global_load_async_to_lds_b128) && \
    __has_builtin(__builtin_amdgcn_global_store_async_from_lds_b128) && \
    __has_builtin(__builtin_amdgcn_s_wait_asynccnt)
#define HAVE_ASYNC_LDS 1
#else
#define HAVE_ASYNC_LDS 0
#endif

#if defined(__AMDGCN__) && __has_builtin(__builtin_amdgcn_tensor_load_to_lds) && \
    __has_builtin(__builtin_amdgcn_s_wait_tensorcnt)
#define HAVE_TDM 1
#else
#define HAVE_TDM 0
#endif

#if defined(__AMDGCN__)
typedef __attribute__((address_space(1))) v4i gv4i;   // global
typedef __attribute__((address_space(3))) v4i lv4i;   // LDS
#define GAS(p) ((gv4i*)(void*)(p))
#define LAS(p) ((lv4i*)(void*)(p))

__device__ __forceinline__ unsigned lds_byte_offset(void* p) {
  // generic -> LDS addrspace cast; AS3 pointer value is the byte offset into LDS
  return (unsigned)(uintptr_t)(__attribute__((address_space(3))) char*)p;
}
#endif

// ---------------- kernel A: fold the two dense layers ----------------
// M[64x64] (padded) = Wp[49,128] @ Wb[128,49] ; base[n][64] = (bp+pos_emb[n])@Wb + bb
__global__ __launch_bounds__(256) void kprep(const float* __restrict__ Wp,
                                             const float* __restrict__ bp,
                                             const float* __restrict__ pe,
                                             const float* __restrict__ Wb,
                                             const float* __restrict__ bb,
                                             float* __restrict__ M,
                                             float* __restrict__ base) {
  int t = blockIdx.x * blockDim.x + threadIdx.x;
  int nthr = gridDim.x * blockDim.x;
  for (int idx = t; idx < 64 * 64; idx += nthr) {
    int k = idx >> 6, n = idx & 63;
    float s = 0.f;
    if (k < PATCHV && n < PATCHV)
      for (int e = 0; e < 128; ++e) s += Wp[k * 128 + e] * Wb[e * PATCHV + n];
    M[idx] = s;
  }
  for (int idx = t; idx < NPq * 64; idx += nthr) {
    int n = idx >> 6, col = idx & 63;
    float s = 0.f;
    if (col < PATCHV) {
      for (int e = 0; e < 128; ++e) s += (bp[e] + pe[n * 128 + e]) * Wb[e * PATCHV + col];
      s += bb[col];
    }
    base[idx] = s;
  }
}

// ---------------- kernel B: similarity + top-10 + transposed extraction ----------------
__global__ __launch_bounds__(64) void ksim(const float* __restrict__ img,
                                           float* __restrict__ patches,
                                           int* __restrict__ xi,
                                           int* __restrict__ yi) {
  __shared__ float sWin[WIN * WIN];
  __shared__ float sRef[PATCHV];
  __shared__ float sSim[Lc * Lc];
  __shared__ int   sIdx[NPq];
  const int t  = threadIdx.x;
  const int pb = blockIdx.x;
  const int pos = pb >> 5;       // /32
  const int bc  = pb & 31;
  const int px = pos / NX, py = pos % NX;   // x outer, y inner
  const int x = px * STRIDE, y = py * STRIDE;
  const int xb0 = (x - HALF > 0) ? (x - HALF) : 0;
  const int yb0 = (y - HALF > 0) ? (y - HALF) : 0;
  const int jmax = x + 6 - xb0;  // winW - P
  const int imax = y + 6 - yb0;  // winH - P
  const float* plane = img + (size_t)bc * PLANE;

  for (int e = t; e < WIN * WIN; e += 64) {
    int r = e / WIN, c = e % WIN;
    sWin[e] = plane[(yb0 + r) * Wc + xb0 + c];
  }
  for (int e = t; e < PATCHV; e += 64) {
    int a = e / Pz, b = e % Pz;
    sRef[e] = plane[(y + a) * Wc + x + b];
  }
  __syncthreads();

  const float NEG = -__builtin_inff();
  for (int e = t; e < Lc * Lc; e += 64) {
    int i = e / Lc, j = e % Lc;
    float s = 0.f;
#pragma unroll
    for (int a = 0; a < Pz; ++a)
#pragma unroll
      for (int b = 0; b < Pz; ++b)
        s += sWin[(i + a) * WIN + (j + b)] * sRef[a * Pz + b];
    if (i > imax || j > jmax) s = NEG;
    sSim[e] = s;
  }
  __syncthreads();

  if (t == 0) {  // top-10, lax.top_k tie-break (smallest index on equal values)
    for (int k = 0; k < NPq; ++k) {
      float best = NEG; int bi = 0;
      for (int q = 0; q < Lc * Lc; ++q) {
        float v = sSim[q];
        if (v > best) { best = v; bi = q; }
      }
      sIdx[k] = bi;
      sSim[bi] = NEG;
    }
  }
  __syncthreads();

  const size_t gb = ((size_t)pos * BCc + bc) * NPq;
  for (int e = t; e < NPq * PATCHV; e += 64) {
    int k = e / PATCHV, q = e % PATCHV;
    int i = q / Pz, j = q % Pz;
    int a = sIdx[k] / Lc, b = sIdx[k] % Lc;
    patches[(gb + k) * PATCHV + q] = sWin[(a + j) * WIN + (b + i)]; // transposed patch
  }
  if (t < NPq) {
    int a = sIdx[t] / Lc, b = sIdx[t] % Lc;
    xi[gb + t] = a + xb0;  // source bug kept: row offset + x window start
    yi[gb + t] = b + yb0;
  }
}

// ---------------- kernel C: den = patches @ M + base  (WMMA f32 16x16x4, TDM staging) ----------------
__global__ __launch_bounds__(32) void kpost(const float* patchesIn,
                                            float* denOut,   // same buffer (in-place, tile-owned)
                                            const float* __restrict__ Mg,
                                            const float* __restrict__ base) {
  __shared__ float As[16 * PATCHV];   // packed rows of 49 (TDM writes rows contiguously)
  __shared__ float Ms[64 * 64];
  const int lane = threadIdx.x;
  const int g0 = blockIdx.x * 16;

#if HAVE_TDM
  {
    // --- D# for the A tile: 2D, data_size=4B, tile 49x16, row stride 49 elements ---
    unsigned long long ga = (unsigned long long)(uintptr_t)(patchesIn + (size_t)g0 * PATCHV);
    u32x4 d0;
    d0[0] = 1u;                                              // count=1, user descriptor
    d0[1] = lds_byte_offset(&As[0]);                         // lds_addr
    d0[2] = (unsigned)(ga & 0xFFFFFFFFu);                    // global_addr[31:0]
    d0[3] = (unsigned)((ga >> 32) & 0x1FFFFFFu) | (2u << 30);// global_addr[56:32] | type=2
    i32x8 d1;
    d1[0] = (int)(2u << 16);          // workgroup_mask=0, data_size=2 (4B)
    d1[1] = (int)((unsigned)PATCHV << 16);   // tensor_dim0[15:0]
    d1[2] = (int)(16u << 16);         // tensor_dim0 hi | tensor_dim1=16
    d1[3] = (int)((unsigned)PATCHV << 16);   // tensor_dim1 hi | tile_dim0=49
    d1[4] = 16;                       // tile_dim1=16, tile_dim2=0
    d1[5] = PATCHV;                   // tensor_dim0_stride low
    d1[6] = 0; d1[7] = 0;
    i32x4 z4 = {0, 0, 0, 0};
    i32x8 z8 = {0, 0, 0, 0, 0, 0, 0, 0};
    __builtin_amdgcn_tensor_load_to_lds(d0, d1, z4, z4, z8, 0);

    // --- D# for M: 1D copy of 4096 contiguous f32 ---
    unsigned long long gm = (unsigned long long)(uintptr_t)Mg;
    u32x4 m0;
    m0[0] = 1u;
    m0[1] = lds_byte_offset(&Ms[0]);
    m0[2] = (unsigned)(gm & 0xFFFFFFFFu);
    m0[3] = (unsigned)((gm >> 32) & 0x1FFFFFFu) | (2u << 30);
    i32x8 m1;
    m1[0] = (int)(2u << 16);
    m1[1] = (int)(4096u << 16);       // tensor_dim0 = 4096
    m1[2] = 0;
    m1[3] = (int)(4096u << 16);       // tile_dim0 = 4096
    m1[4] = 0;                        // tile_dim1 unused -> 1D
    m1[5] = 4096;
    m1[6] = 0; m1[7] = 0;
    __builtin_amdgcn_tensor_load_to_lds(m0, m1, z4, z4, z8, 0);

    __builtin_amdgcn_s_wait_tensorcnt(0);
  }
#else
  for (int r = 0; r < 16; ++r)
    for (int c = lane; c < PATCHV; c += 32)
      As[r * PATCHV + c] = patchesIn[(size_t)(g0 + r) * PATCHV + c];
  for (int i = lane; i < 64 * 64; i += 32) Ms[i] = Mg[i];
#endif
  __syncthreads();

  const int mrow  = lane & 15;
  const int khalf = (lane >> 4) * 2;      // lanes 0-15 hold K={0,1}, lanes 16-31 K={2,3}
  v2f afrag[16];
#pragma unroll
  for (int k = 0; k < 16; ++k) {
    int c0 = 4 * k + khalf;
    afrag[k].x = (c0 < PATCHV)     ? As[mrow * PATCHV + c0]     : 0.f;
    afrag[k].y = (c0 + 1 < PATCHV) ? As[mrow * PATCHV + c0 + 1] : 0.f;
  }

  const int rowoff = (lane >> 4) * 8;     // D: VGPR r -> M=r (lanes 0-15) / M=r+8 (16-31)
#pragma unroll 1
  for (int nt = 0; nt < 4; ++nt) {
    v8f acc = {};
    const int ncol = nt * 16 + mrow;
#pragma unroll
    for (int k = 0; k < 16; ++k) {
      v2f b;
      b.x = Ms[(4 * k + khalf) * 64 + ncol];
      b.y = Ms[(4 * k + khalf + 1) * 64 + ncol];
      acc = __builtin_amdgcn_wmma_f32_16x16x4_f32(false, afrag[k], false, b,
                                                  (short)0, acc, false, false);
    }
#pragma unroll
    for (int r = 0; r < 8; ++r) {
      int row = r + rowoff;
      int g = g0 + row;
      if (ncol < PATCHV)
        denOut[(size_t)g * PATCHV + ncol] = acc[r] + base[(g % NPq) * 64 + ncol];
    }
  }
}

// ---------------- kernel D: sequential scatter-blend, whole plane in LDS ----------------
__global__ __launch_bounds__(256) void kscatter(const float* __restrict__ img,
                                                const float* __restrict__ den,
                                                const int* __restrict__ xi,
                                                const int* __restrict__ yi,
                                                float* __restrict__ out) {
  extern __shared__ float lds[];
  float* im  = lds;            // 25600 floats
  float* cnt = lds + PLANE;    // 25600 floats
  const int plane = blockIdx.x;
  const int t = threadIdx.x;
  const float* src = img + (size_t)plane * PLANE;

#if HAVE_ASYNC_LDS
  for (int e = t * 4; e < PLANE; e += 256 * 4)
    __builtin_amdgcn_global_load_async_to_lds_b128(GAS(src + e), LAS(im + e), 0, 0);
#else
  for (int e = t; e < PLANE; e += 256) im[e] = src[e];
#endif
  for (int e = t; e < PLANE; e += 256) cnt[e] = 1.0f;
#if HAVE_ASYNC_LDS
  __builtin_amdgcn_s_wait_asynccnt(0);
#endif
  __syncthreads();

  if (t < 32) {                // single wave32 runs the dependency chain (in-order LDS)
    const int lane = t;
    const bool has1 = (lane + 32) < PATCHV;
    const int e0 = lane, e1 = lane + 32;
    const int i0 = e0 / Pz, j0 = e0 % Pz;
    const int i1 = e1 / Pz, j1 = e1 % Pz;
    for (int step = 0; step < NSTEP; ++step) {
      const int pos = step / NPq, n = step % NPq;
      const size_t idx = ((size_t)pos * BCc + plane) * NPq + n;
      const int X = xi[idx];
      const int Y = yi[idx];
      const float* d = den + idx * PATCHV;
      if (step + 4 < NSTEP) {  // gfx1250 global_prefetch for the den stream
        size_t idx2 = ((size_t)((step + 4) / NPq) * BCc + plane) * NPq + ((step + 4) % NPq);
        __builtin_prefetch(den + idx2 * PATCHV + lane, 0, 0);
      }
      // --- read all of c,v,d first (reference reads counters pre-increment) ---
      int o0 = (Y + i0) * Wc + (X + j0);
      float c0 = cnt[o0], v0 = im[o0], d0 = d[e0];
      int o1 = 0; float c1 = 0.f, v1 = 0.f, d1 = 0.f;
      if (has1) { o1 = (Y + i1) * Wc + (X + j1); c1 = cnt[o1]; v1 = im[o1]; d1 = d[e1]; }
      asm volatile("" ::: "memory");
      // --- blend writes ---
      im[o0] = (v0 * c0 + d0) / (c0 + 1.0f);
      if (has1) im[o1] = (v1 * c1 + d1) / (c1 + 1.0f);
      asm volatile("" ::: "memory");
      // --- counter increment at the TRANSPOSED slice (source bug kept) ---
      int p0 = (X + i0) * Wc + (Y + j0);
      float cp0 = cnt[p0];
      cnt[p0] = cp0 + 1.0f;
      if (has1) {
        int p1 = (X + i1) * Wc + (Y + j1);
        float cp1 = cnt[p1];
        cnt[p1] = cp1 + 1.0f;
      }
    }
  }
  __syncthreads();

  float* dst = out + (size_t)plane * PLANE;
#if HAVE_ASYNC_LDS
  for (int e = t * 4; e < PLANE; e += 256 * 4)
    __builtin_amdgcn_global_store_async_from_lds_b128(GAS(dst + e), LAS(im + e), 0, 0);
  __builtin_amdgcn_s_wait_asynccnt(0);
#else
  for (int e = t; e < PLANE; e += 256) dst[e] = im[e];
#endif
}

// ---------------- launch ----------------
extern "C" void kernel_launch(void* const* d_in, const int* in_sizes, int n_in,
                              void* d_out, int out_size, void* d_ws, size_t ws_size,
                              hipStream_t stream) {
  const float* images = (const float*)d_in[0];
  const float* Wp     = (const float*)d_in[1];
  const float* bp     = (const float*)d_in[2];
  const float* pe     = (const float*)d_in[3];
  const float* Wb     = (const float*)d_in[4];
  const float* bb     = (const float*)d_in[5];
  float* out = (float*)d_out;

  float* ws      = (float*)d_ws;
  float* M       = ws;                                   // 64*64
  float* base    = ws + 4096;                            // 10*64
  float* patches = ws + 8192;                            // NPATCH*49 (den in-place)
  int*   xi      = (int*)(ws + 8192 + (size_t)NPATCH * PATCHV);
  int*   yi      = xi + NPATCH;

  kprep<<<16, 256, 0, stream>>>(Wp, bp, pe, Wb, bb, M, base);
  ksim<<<NPOS * BCc, 64, 0, stream>>>(images, patches, xi, yi);
  kpost<<<TILES, 32, 0, stream>>>(patches, patches, M, base);
  kscatter<<<BCc, 256, 2 * PLANE * sizeof(float), stream>>>(images, patches, xi, yi, out);
}